// PubMedGAT_56796647522840
// MI455X (gfx1250) — compile-verified
//
#include <hip/hip_runtime.h>

#define NN 50000
#define NE 800000
#define INDIM 512
#define HEADS 4
#define HID 64
#define OUTD 64
#define HF (HEADS * HID)   // 256
#define NEG_SLOPE 0.2f

typedef __attribute__((ext_vector_type(2))) float v2f;
typedef __attribute__((ext_vector_type(8))) float v8f;

#if __has_builtin(__builtin_amdgcn_global_load_async_to_lds_b128)
#define HAVE_ASYNC_LDS 1
// Builtin expects: (v4i addrspace(1)*, v4i addrspace(3)*, imm offset, imm cpol)
typedef int v4i __attribute__((vector_size(16)));
typedef __attribute__((address_space(1))) v4i as1_v4i;
typedef __attribute__((address_space(3))) v4i as3_v4i;
#else
#define HAVE_ASYNC_LDS 0
#endif

__device__ __forceinline__ void wait_async_lds() {
#if HAVE_ASYNC_LDS
#if __has_builtin(__builtin_amdgcn_s_wait_asynccnt)
  __builtin_amdgcn_s_wait_asynccnt(0);
#else
  asm volatile("s_wait_asynccnt 0x0" ::: "memory");
#endif
#endif
}

// Monotonic order-preserving float<->uint key for atomic max.
#define KEY_NEG_INF 0x007fffffu
__device__ __forceinline__ unsigned f2key(float f) {
  unsigned u = __float_as_uint(f);
  return (u & 0x80000000u) ? ~u : (u | 0x80000000u);
}
__device__ __forceinline__ float key2f(unsigned k) {
  unsigned u = (k & 0x80000000u) ? (k & 0x7fffffffu) : ~k;
  return __uint_as_float(u);
}

// ------------------------------------------------------------------
// fp32 GEMM via V_WMMA_F32_16X16X4_F32.  C[M,N] = A[M,K]*B[K,N].
// Requires N % 64 == 0, K % 32 == 0; M guarded.
// Block = 256 threads (8 waves); block tile 128x64; wave tile 16x64.
// A tile staged via async global->LDS (when available); B tile stored
// K-pair-interleaved so each B fragment is one aligned 8B LDS load.
// ------------------------------------------------------------------
#define MT 128
#define NT 64
#define KC 32
#define AS_STRIDE (KC + 4)   // +4 floats: keeps 16B row alignment, spreads banks
#define BS_PITCH (2 * NT)    // floats per interleaved K-pair row

__global__ __launch_bounds__(256) void gemm_f32_wmma(const float* __restrict__ A,
                                                     const float* __restrict__ B,
                                                     float* __restrict__ C,
                                                     int M, int N, int K) {
  __shared__ float As[MT * AS_STRIDE];
  __shared__ float Bs[(KC / 2) * BS_PITCH];

  const int tid   = threadIdx.x;
  const int wave  = tid >> 5;
  const int lane  = tid & 31;
  const int lhalf = lane >> 4;   // 0: lanes 0-15, 1: lanes 16-31
  const int l16   = lane & 15;
  const int m0    = blockIdx.x * MT;
  const int n0    = blockIdx.y * NT;

  v8f acc[4] = {};

  for (int k0 = 0; k0 < K; k0 += KC) {
    // ---- Stage A tile: 128x32 floats = 1024 float4 slots, 4 per thread. ----
#pragma unroll
    for (int i = 0; i < 4; ++i) {
      int s   = tid + i * 256;
      int row = s >> 3;          // 8 float4 per row
      int cg  = s & 7;
      int grow = m0 + row;
      float* lptr = &As[row * AS_STRIDE + cg * 4];
      if (grow < M) {
        const float* gptr = A + (size_t)grow * K + k0 + cg * 4;
#if HAVE_ASYNC_LDS
        __builtin_amdgcn_global_load_async_to_lds_b128(
            (as1_v4i*)gptr, (as3_v4i*)lptr, 0, 0);
#else
        *(float4*)lptr = *(const float4*)gptr;
#endif
      } else {
        *(float4*)lptr = make_float4(0.f, 0.f, 0.f, 0.f);
      }
    }
    // ---- Stage B tile: 32x64 floats, K-pair interleaved; 1 slot/thread. ----
    {
      int p  = tid >> 4;        // K pair-row 0..15
      int cg = tid & 15;        // column group (4 cols)
      const float* b0 = B + (size_t)(k0 + 2 * p) * N + n0 + cg * 4;
      float4 r0 = *(const float4*)b0;
      float4 r1 = *(const float4*)(b0 + N);
      float* dp = &Bs[p * BS_PITCH + cg * 8];
      *(float2*)(dp + 0) = make_float2(r0.x, r1.x);
      *(float2*)(dp + 2) = make_float2(r0.y, r1.y);
      *(float2*)(dp + 4) = make_float2(r0.z, r1.z);
      *(float2*)(dp + 6) = make_float2(r0.w, r1.w);
    }
    wait_async_lds();
    __syncthreads();

    const int ar = wave * 16 + l16;      // A row for this lane (M = lane%16)
#pragma unroll
    for (int kk = 0; kk < KC; kk += 4) {
      // A 16x4 fp32 fragment: lanes 0-15 hold K={kk,kk+1}, lanes 16-31 K={kk+2,kk+3}
      int ka = kk + lhalf * 2;           // even
      v2f a = *(const v2f*)&As[ar * AS_STRIDE + ka];
#pragma unroll
      for (int nt = 0; nt < 4; ++nt) {
        // B 4x16 fragment: pair (ka, ka+1) contiguous in interleaved layout
        v2f b = *(const v2f*)&Bs[(ka >> 1) * BS_PITCH + 2 * (nt * 16 + l16)];
        acc[nt] = __builtin_amdgcn_wmma_f32_16x16x4_f32(
            false, a, false, b, (short)0, acc[nt], false, false);
      }
    }
    __syncthreads();
  }

  // C/D layout: VGPR v -> row v (lanes 0-15) / v+8 (lanes 16-31), col = lane%16
#pragma unroll
  for (int nt = 0; nt < 4; ++nt) {
    int col = n0 + nt * 16 + l16;
#pragma unroll
    for (int v = 0; v < 8; ++v) {
      int row = m0 + wave * 16 + v + lhalf * 8;
      if (row < M) C[(size_t)row * N + col] = acc[nt][v];
    }
  }
}

// ------------------------------------------------------------------
// Layer 1 attention kernels
// ------------------------------------------------------------------
__global__ void l1_node_prep(const float* __restrict__ f1,
                             const float* __restrict__ al1,
                             const float* __restrict__ ar1,
                             float* __restrict__ el1, float* __restrict__ er1,
                             unsigned* __restrict__ m1, float* __restrict__ den1,
                             float* __restrict__ agg1) {
  int t = blockIdx.x * blockDim.x + threadIdx.x;
  if (t >= NN * HEADS) return;
  int n = t >> 2, h = t & 3;
  const float* fr  = f1 + (size_t)n * HF + h * HID;
  const float* alp = al1 + h * HID;
  const float* arp = ar1 + h * HID;
  float* ag = agg1 + (size_t)n * HF + h * HID;
  float el = 0.f, er = 0.f;
#pragma unroll 4
  for (int d = 0; d < HID; ++d) {
    float fv = fr[d];
    el += fv * alp[d];
    er += fv * arp[d];
    ag[d] = 0.f;
  }
  el1[t] = el;
  er1[t] = er;
  m1[t]  = KEY_NEG_INF;
  den1[t] = 0.f;
}

__global__ void l1_edge_max(const int* __restrict__ src, const int* __restrict__ dst,
                            const float* __restrict__ el1, const float* __restrict__ er1,
                            unsigned* __restrict__ m1) {
  int t = blockIdx.x * blockDim.x + threadIdx.x;
  if (t >= NE * HEADS) return;
  int e = t >> 2, h = t & 3;
  int s = src[e], d = dst[e];
  float x = el1[s * HEADS + h] + er1[d * HEADS + h];
  x = x > 0.f ? x : NEG_SLOPE * x;
  atomicMax(&m1[d * HEADS + h], f2key(x));
}

// One 256-thread block per edge: coalesced gather of f1[src] row,
// un-normalized exp-weighted scatter into agg1[dst].
__global__ __launch_bounds__(256) void l1_edge_agg(
    const int* __restrict__ src, const int* __restrict__ dst,
    const float* __restrict__ el1, const float* __restrict__ er1,
    const unsigned* __restrict__ m1, const float* __restrict__ f1,
    float* __restrict__ agg1, float* __restrict__ den1) {
  int e = blockIdx.x;
  int c = threadIdx.x;          // channel 0..255
  int h = c >> 6;
  int s = src[e], d = dst[e];
  float x = el1[s * HEADS + h] + er1[d * HEADS + h];
  x = x > 0.f ? x : NEG_SLOPE * x;
  float ex = __expf(x - key2f(m1[d * HEADS + h]));
  atomicAdd(&agg1[(size_t)d * HF + c], ex * f1[(size_t)s * HF + c]);
  if ((c & 63) == 0) atomicAdd(&den1[d * HEADS + h], ex);
}

__global__ void l1_finalize(const float* __restrict__ agg1,
                            const float* __restrict__ den1,
                            const float* __restrict__ b1,
                            float* __restrict__ h1) {
  long t = (long)blockIdx.x * blockDim.x + threadIdx.x;
  if (t >= (long)NN * HF) return;
  int c = (int)(t & (HF - 1));
  int n = (int)(t >> 8);
  int h = c >> 6;
  float den = den1[n * HEADS + h];
  float v = (den > 0.f) ? (agg1[t] / den) : 0.f;   // no in-edges -> bias only
  v += b1[c];
  h1[t] = v > 0.f ? v : (__expf(v) - 1.f);         // ELU
}

// ------------------------------------------------------------------
// Layer 2 attention kernels (H = 1, D = 64); agg accumulates into d_out.
// ------------------------------------------------------------------
__global__ void l2_node_prep(const float* __restrict__ f2,
                             const float* __restrict__ al2,
                             const float* __restrict__ ar2,
                             float* __restrict__ el2, float* __restrict__ er2,
                             unsigned* __restrict__ m2, float* __restrict__ den2,
                             float* __restrict__ out) {
  int n = blockIdx.x * blockDim.x + threadIdx.x;
  if (n >= NN) return;
  const float* fr = f2 + (size_t)n * OUTD;
  float* op = out + (size_t)n * OUTD;
  float el = 0.f, er = 0.f;
#pragma unroll 4
  for (int d = 0; d < OUTD; ++d) {
    float fv = fr[d];
    el += fv * al2[d];
    er += fv * ar2[d];
    op[d] = 0.f;
  }
  el2[n] = el;
  er2[n] = er;
  m2[n]  = KEY_NEG_INF;
  den2[n] = 0.f;
}

__global__ void l2_edge_max(const int* __restrict__ src, const int* __restrict__ dst,
                            const float* __restrict__ el2, const float* __restrict__ er2,
                            unsigned* __restrict__ m2) {
  int e = blockIdx.x * blockDim.x + threadIdx.x;
  if (e >= NE) return;
  int s = src[e], d = dst[e];
  float x = el2[s] + er2[d];
  x = x > 0.f ? x : NEG_SLOPE * x;
  atomicMax(&m2[d], f2key(x));
}

__global__ __launch_bounds__(64) void l2_edge_agg(
    const int* __restrict__ src, const int* __restrict__ dst,
    const float* __restrict__ el2, const float* __restrict__ er2,
    const unsigned* __restrict__ m2, const float* __restrict__ f2,
    float* __restrict__ out, float* __restrict__ den2) {
  int e = blockIdx.x;
  int c = threadIdx.x;          // 0..63
  int s = src[e], d = dst[e];
  float x = el2[s] + er2[d];
  x = x > 0.f ? x : NEG_SLOPE * x;
  float ex = __expf(x - key2f(m2[d]));
  atomicAdd(&out[(size_t)d * OUTD + c], ex * f2[(size_t)s * OUTD + c]);
  if (c == 0) atomicAdd(&den2[d], ex);
}

__global__ void l2_finalize(const float* __restrict__ den2,
                            const float* __restrict__ b2,
                            float* __restrict__ out) {
  long t = (long)blockIdx.x * blockDim.x + threadIdx.x;
  if (t >= (long)NN * OUTD) return;
  int n = (int)(t >> 6);
  int c = (int)(t & (OUTD - 1));
  float den = den2[n];
  float v = (den > 0.f) ? (out[t] / den) : 0.f;
  out[t] = v + b2[c];
}

// ------------------------------------------------------------------
extern "C" void kernel_launch(void* const* d_in, const int* in_sizes, int n_in,
                              void* d_out, int out_size, void* d_ws, size_t ws_size,
                              hipStream_t stream) {
  const float* feat = (const float*)d_in[0];   // [NN, 512]
  const float* W1   = (const float*)d_in[1];   // [512, 256]
  const float* al1  = (const float*)d_in[2];   // [4, 64]
  const float* ar1  = (const float*)d_in[3];
  const float* b1   = (const float*)d_in[4];
  const float* W2   = (const float*)d_in[5];   // [256, 64]
  const float* al2  = (const float*)d_in[6];   // [1, 64]
  const float* ar2  = (const float*)d_in[7];
  const float* b2   = (const float*)d_in[8];
  const int*   src  = (const int*)d_in[9];
  const int*   dst  = (const int*)d_in[10];
  float* out = (float*)d_out;                  // [NN, 1, 64]

  // Workspace carve-up (fp32 elements). f2 reuses agg1's region; h1 reuses f1.
  float*    f1   = (float*)d_ws;                       // NN*256
  float*    agg1 = f1 + (size_t)NN * HF;               // NN*256 (later: f2)
  float*    el1  = agg1 + (size_t)NN * HF;             // NN*4
  float*    er1  = el1 + (size_t)NN * HEADS;           // NN*4
  unsigned* m1   = (unsigned*)(er1 + (size_t)NN * HEADS);  // NN*4
  float*    den1 = (float*)(m1 + (size_t)NN * HEADS);  // NN*4
  float*    el2  = den1 + (size_t)NN * HEADS;          // NN
  float*    er2  = el2 + NN;                           // NN
  unsigned* m2   = (unsigned*)(er2 + NN);              // NN
  float*    den2 = (float*)(m2 + NN);                  // NN
  float*    h1 = f1;
  float*    f2 = agg1;

  // ---- Layer 1 ----
  {
    dim3 grid((NN + MT - 1) / MT, HF / NT);
    gemm_f32_wmma<<<grid, 256, 0, stream>>>(feat, W1, f1, NN, HF, INDIM);
  }
  l1_node_prep<<<(NN * HEADS + 255) / 256, 256, 0, stream>>>(f1, al1, ar1, el1, er1, m1, den1, agg1);
  l1_edge_max<<<(NE * HEADS + 255) / 256, 256, 0, stream>>>(src, dst, el1, er1, m1);
  l1_edge_agg<<<NE, 256, 0, stream>>>(src, dst, el1, er1, m1, f1, agg1, den1);
  l1_finalize<<<((long)NN * HF + 255) / 256, 256, 0, stream>>>(agg1, den1, b1, h1);

  // ---- Layer 2 ----
  {
    dim3 grid((NN + MT - 1) / MT, OUTD / NT);
    gemm_f32_wmma<<<grid, 256, 0, stream>>>(h1, W2, f2, NN, OUTD, HF);
  }
  l2_node_prep<<<(NN + 255) / 256, 256, 0, stream>>>(f2, al2, ar2, el2, er2, m2, den2, out);
  l2_edge_max<<<(NE + 255) / 256, 256, 0, stream>>>(src, dst, el2, er2, m2);
  l2_edge_agg<<<NE, 64, 0, stream>>>(src, dst, el2, er2, m2, f2, out, den2);
  l2_finalize<<<((long)NN * OUTD + 255) / 256, 256, 0, stream>>>(den2, b2, out);
}